// ROIMerge_55722905698379
// MI455X (gfx1250) — compile-verified
//
#include <hip/hip_runtime.h>
#include <stdint.h>

typedef uint32_t u32;
typedef unsigned long long u64;

typedef uint32_t __attribute__((ext_vector_type(4))) v4u;
typedef int      __attribute__((ext_vector_type(4))) v4i;
typedef int      __attribute__((ext_vector_type(8))) v8i;

#define IOU_THRESH 0.5f
#define NMAX 8192

// ---------------------------------------------------------------------------
// TDM: issue a 1D tensor_load_to_lds copying `nelem` floats from gptr into the
// workgroup's LDS at byte offset lds_byte_off. D# layout per CDNA5 ISA ch.8.
// ---------------------------------------------------------------------------
__device__ __forceinline__ void tdm_load_row(const float* gptr,
                                             uint32_t lds_byte_off,
                                             int nelem) {
  uint64_t ga = (uint64_t)(uintptr_t)gptr;

  v4u g0;
  g0[0] = 1u;                                              // count=1 (valid), user mode
  g0[1] = lds_byte_off;                                    // lds_addr [63:32]
  g0[2] = (uint32_t)(ga & 0xFFFFFFFFu);                    // global_addr [95:64]
  g0[3] = (uint32_t)((ga >> 32) & 0x01FFFFFFu)             // global_addr [120:96]
          | 0x80000000u;                                   // type=2 ("image") [127:126]

  v8i g1;
  g1[0] = (int)(2u << 16);                                 // workgroup_mask=0, data_size=2 (4B)
  g1[1] = (int)(((u32)nelem & 0xFFFFu) << 16);             // tensor_dim0 lo16 @ [31:16]
  g1[2] = (int)((((u32)nelem >> 16) & 0xFFFFu)             // tensor_dim0 hi16
          | (1u << 16));                                   // tensor_dim1 = 1
  g1[3] = (int)(((u32)nelem & 0xFFFFu) << 16);             // tile_dim0 = nelem @ [127:112]
  g1[4] = 1;                                               // tile_dim1 = 1, tile_dim2 = 0
  g1[5] = nelem;                                           // tensor_dim0_stride lo32
  g1[6] = 0;                                               // strides hi bits
  g1[7] = 0;

  v4i gz = {0, 0, 0, 0};                                   // groups 2/3 unused (<=2D)
#if defined(__clang_major__) && (__clang_major__ >= 23)
  v8i gz8 = {0, 0, 0, 0, 0, 0, 0, 0};
  __builtin_amdgcn_tensor_load_to_lds(g0, g1, gz, gz, gz8, 0);
#else
  __builtin_amdgcn_tensor_load_to_lds(g0, g1, gz, gz, 0);
#endif
}

__device__ __forceinline__ u64 shfl_xor_u64(u64 v, int mask) {
  u32 lo = (u32)v;
  u32 hi = (u32)(v >> 32);
  lo = __shfl_xor(lo, mask, 32);
  hi = __shfl_xor(hi, mask, 32);
  return ((u64)hi << 32) | (u64)lo;
}

// ---------------------------------------------------------------------------
// Kernel 1: leader[i] = argmax_j (J[i,j] > thresh ? S[j] : -inf).
// One block per row; TDM streams the 32KB row into LDS; wave32 key-reduction.
// ---------------------------------------------------------------------------
__global__ __launch_bounds__(256) void leaders_kernel(
    const float* __restrict__ S, const float* __restrict__ J,
    int* __restrict__ leader, int N) {
  __shared__ float row[NMAX];
  __shared__ u64 red;

  const int i = blockIdx.x;
  if (threadIdx.x == 0) red = 0ull;

  if (threadIdx.x < 32) {  // wave 0 only: TDM issue (EXEC ignored by TDM)
    tdm_load_row(J + (size_t)i * (size_t)N, (uint32_t)(uintptr_t)&row[0], N);
    __builtin_amdgcn_s_wait_tensorcnt(0);
  }
  __syncthreads();

  const float4* __restrict__ r4 = (const float4*)row;
  const float4* __restrict__ s4 = (const float4*)S;
  float bestS = -1.0f;
  int bestI = 0;
  const int nq = N >> 2;
  for (int c = threadIdx.x; c < nq; c += blockDim.x) {
    float4 jv = r4[c];          // ds_load_b128 from LDS
    float4 sv = s4[c];          // global_load_b128, L2/WGP$-resident (32KB)
    int j0 = c << 2;
    if (jv.x > IOU_THRESH && sv.x > bestS) { bestS = sv.x; bestI = j0;     }
    if (jv.y > IOU_THRESH && sv.y > bestS) { bestS = sv.y; bestI = j0 + 1; }
    if (jv.z > IOU_THRESH && sv.z > bestS) { bestS = sv.z; bestI = j0 + 2; }
    if (jv.w > IOU_THRESH && sv.w > bestS) { bestS = sv.w; bestI = j0 + 3; }
  }

  // key: (float bits of S) high, inverted index low -> max key == argmax with
  // first-occurrence tie-break, matching jnp.argmax exactly.
  u64 key = (bestS < 0.0f)
                ? 0ull
                : (((u64)__float_as_uint(bestS) << 32) |
                   (u64)(0xFFFFFFFFu - (u32)bestI));

  for (int off = 16; off > 0; off >>= 1) {
    u64 o = shfl_xor_u64(key, off);
    key = (o > key) ? o : key;
  }
  if ((threadIdx.x & 31) == 0) atomicMax(&red, key);
  __syncthreads();
  if (threadIdx.x == 0) leader[i] = (int)(0xFFFFFFFFu - (u32)red);
}

// ---------------------------------------------------------------------------
// Kernel 2: segment sums + counts via atomics (buffers pre-zeroed).
// ---------------------------------------------------------------------------
__global__ __launch_bounds__(128) void accum_kernel(
    const float* __restrict__ C, const float* __restrict__ D,
    const int* __restrict__ leader, float* __restrict__ counts,
    float* __restrict__ sumC, float* __restrict__ sumD, int N, int K) {
  const int i = blockIdx.x;
  const int l = leader[i];
  if (threadIdx.x == 0) atomicAdd(&counts[l], 1.0f);
  for (int k = threadIdx.x; k < K; k += blockDim.x) {
    atomicAdd(&sumC[(size_t)l * K + k], C[(size_t)i * K + k]);
    atomicAdd(&sumD[(size_t)l * K + k], D[(size_t)i * K + k]);
  }
}

// Kernel 3: counts -> 1/max(counts,1)
__global__ void invcnt_kernel(float* __restrict__ counts, int N) {
  int i = blockIdx.x * blockDim.x + threadIdx.x;
  if (i < N) {
    float c = counts[i];
    counts[i] = 1.0f / (c > 1.0f ? c : 1.0f);
  }
}

// ---------------------------------------------------------------------------
// Kernel 4: broadcast clique means back to every member row.
// ---------------------------------------------------------------------------
__global__ __launch_bounds__(256) void gather_kernel(
    const int* __restrict__ leader, const float* __restrict__ counts,
    const float* __restrict__ sumC, const float* __restrict__ sumD,
    float* __restrict__ outC, float* __restrict__ outD, int N, int K) {
  const int total = N * K;
  for (int idx = blockIdx.x * blockDim.x + threadIdx.x; idx < total;
       idx += gridDim.x * blockDim.x) {
    const int i = idx / K;
    const int k = idx - i * K;
    const int l = leader[i];
    const float inv = counts[l];
    outC[idx] = sumC[(size_t)l * K + k] * inv;
    outD[idx] = sumD[(size_t)l * K + k] * inv;
  }
}

extern "C" void kernel_launch(void* const* d_in, const int* in_sizes, int n_in,
                              void* d_out, int out_size, void* d_ws,
                              size_t ws_size, hipStream_t stream) {
  const float* S = (const float*)d_in[0];
  const float* J = (const float*)d_in[1];
  const float* C = (const float*)d_in[2];
  const float* D = (const float*)d_in[3];
  const int N = in_sizes[0];
  const int K = in_sizes[2] / N;

  // workspace layout: leader[N] | counts[N] | sumC[N*K] | sumD[N*K]
  char* ws = (char*)d_ws;
  int* leader = (int*)ws;      ws += (size_t)N * sizeof(int);
  float* counts = (float*)ws;  ws += (size_t)N * sizeof(float);
  float* sumC = (float*)ws;    ws += (size_t)N * (size_t)K * sizeof(float);
  float* sumD = (float*)ws;

  float* outC = (float*)d_out;
  float* outD = outC + (size_t)N * (size_t)K;

  // zero counts + sums (contiguous)
  hipMemsetAsync(counts, 0,
                 ((size_t)N + 2ull * (size_t)N * (size_t)K) * sizeof(float),
                 stream);

  leaders_kernel<<<N, 256, 0, stream>>>(S, J, leader, N);
  accum_kernel<<<N, 128, 0, stream>>>(C, D, leader, counts, sumC, sumD, N, K);
  invcnt_kernel<<<(N + 255) / 256, 256, 0, stream>>>(counts, N);
  gather_kernel<<<512, 256, 0, stream>>>(leader, counts, sumC, sumD, outC, outD,
                                         N, K);
}